// AE_GNN_61856118997137
// MI455X (gfx1250) — compile-verified
//
#include <hip/hip_runtime.h>
#include <cstdint>
#include <cstddef>

#define N_NODES   100000
#define NUM_EDGES 3200000
#define D         256
#define N_CLASS   40
#define N_LAYERS  4
#define ALPHA_MIX 0.2f
#define BN_EPS    1e-5f
#define LN_EPS    1e-5f
#define NEG_SLOPE 0.3f

typedef __attribute__((ext_vector_type(2))) float v2f;
typedef __attribute__((ext_vector_type(8))) float v8f;
typedef int v4i_vs __attribute__((vector_size(16)));   // matches builtin param type

#if __has_builtin(__builtin_amdgcn_global_load_async_to_lds_b128)
#define HAVE_ASYNC_LDS 1
#else
#define HAVE_ASYNC_LDS 0
#warning "missing __builtin_amdgcn_global_load_async_to_lds_b128 - using plain LDS fill"
#endif

// Relaxed agent-scope fp32 atomic add -> hardware global_atomic_add_f32 (L2 atomic)
__device__ __forceinline__ void atomAddF32(float* p, float v) {
    __hip_atomic_fetch_add(p, v, __ATOMIC_RELAXED, __HIP_MEMORY_SCOPE_AGENT);
}

// ---------------------------------------------------------------- utilities
__global__ void k_zero(float* p, long n) {
    long i = (long)blockIdx.x * blockDim.x + threadIdx.x;
    long st = (long)gridDim.x * blockDim.x;
    for (; i < n; i += st) p[i] = 0.f;
}

// acc[r, c] = bias[c]  (SpMM accumulator starts at the layer bias)
__global__ void k_fill_bias(float* __restrict__ acc, const float* __restrict__ bias) {
    int c = threadIdx.x;
    float b = bias[c];
    long r0 = (long)blockIdx.x * 4;
#pragma unroll
    for (int i = 0; i < 4; ++i) {
        long r = r0 + i;
        if (r < N_NODES) acc[r * D + c] = b;
    }
}

// ---------------------------------------------------------------- WMMA GEMM
// C[M,256] = (aScale * A[M,256]) @ B[256,256] (+ bias[col])
// One wave -> 16x64 output (4 accumulators); A fragment amortized over 4 WMMAs.
__global__ void k_gemm_wmma(const float* __restrict__ A, const float* __restrict__ B,
                            float* __restrict__ Cout, const float* __restrict__ bias,
                            float aScale) {
    int lane  = threadIdx.x;                 // 0..31 (wave32)
    int tileM = blockIdx.x;                  // 16-row strip (6250 strips)
    int ml = lane & 15;                      // M (A) / N (B,C) within a tile
    int kh = (lane >> 4) << 1;               // K sub-offset: 0 or 2
    int nbase = threadIdx.y * 64 + ml;       // this wave covers cols [wy*64, wy*64+63]

    const float* arow = A + (size_t)(tileM * 16 + ml) * D;
    v8f c0 = {}, c1 = {}, c2 = {}, c3 = {};
#pragma unroll 2
    for (int k = 0; k < D; k += 4) {
        v2f a = *(const v2f*)(arow + k + kh);
        a *= aScale;
        const float* b0 = B + (size_t)(k + kh) * D + nbase;
        const float* b1 = B + (size_t)(k + kh + 1) * D + nbase;
        v2f b;
        b.x = b0[0];  b.y = b1[0];
        c0 = __builtin_amdgcn_wmma_f32_16x16x4_f32(false, a, false, b, (short)0, c0, false, false);
        b.x = b0[16]; b.y = b1[16];
        c1 = __builtin_amdgcn_wmma_f32_16x16x4_f32(false, a, false, b, (short)0, c1, false, false);
        b.x = b0[32]; b.y = b1[32];
        c2 = __builtin_amdgcn_wmma_f32_16x16x4_f32(false, a, false, b, (short)0, c2, false, false);
        b.x = b0[48]; b.y = b1[48];
        c3 = __builtin_amdgcn_wmma_f32_16x16x4_f32(false, a, false, b, (short)0, c3, false, false);
    }
    int rbase = tileM * 16 + ((lane >> 4) << 3);   // lanes 16-31 hold M+8
    float bb0 = bias ? bias[nbase]      : 0.f;
    float bb1 = bias ? bias[nbase + 16] : 0.f;
    float bb2 = bias ? bias[nbase + 32] : 0.f;
    float bb3 = bias ? bias[nbase + 48] : 0.f;
#pragma unroll
    for (int r = 0; r < 8; ++r) {
        float* crow = Cout + (size_t)(rbase + r) * D + nbase;
        crow[0]  = c0[r] + bb0;
        crow[16] = c1[r] + bb1;
        crow[32] = c2[r] + bb2;
        crow[48] = c3[r] + bb3;
    }
}

// ---------------------------------------------------------------- SpMM (COO scatter-add)
// 4 edge-groups x 64 lanes; float4 gather, fp32 L2 atomic scatter. 32 edges/block.
__global__ void k_spmm(const int* __restrict__ esrc, const int* __restrict__ edst,
                       const float* __restrict__ ew, const float* __restrict__ sup,
                       float* acc) {
    int t  = threadIdx.x;
    int g  = t >> 6;               // edge subgroup 0..3
    int c4 = (t & 63) << 2;        // starting column (x4 floats)
    long base = (long)blockIdx.x * 32 + g;
#pragma unroll 1
    for (int i = 0; i < 8; ++i) {
        long e = base + (long)i * 4;
        __builtin_prefetch(&esrc[e + 64], 0, 1);   // global_prefetch_b8
        int s = esrc[e];
        int d = edst[e];
        float w = ew[e];
        float4 v = *(const float4*)(sup + (size_t)s * D + c4);
        float* dp = acc + (size_t)d * D + c4;
        atomAddF32(dp + 0, v.x * w);
        atomAddF32(dp + 1, v.y * w);
        atomAddF32(dp + 2, v.z * w);
        atomAddF32(dp + 3, v.w * w);
    }
}

// ---------------------------------------------------------------- BatchNorm stats
__global__ void k_bn_stats(const float* __restrict__ acc, float* stats) {
    int c = threadIdx.x;
    long r0 = (long)blockIdx.x * 125;               // 800 * 125 = 100000
    float s = 0.f, s2 = 0.f;
    for (int i = 0; i < 125; ++i) {
        float v = acc[(r0 + i) * D + c];
        s += v;
        s2 += v * v;
    }
    atomAddF32(&stats[c], s);
    atomAddF32(&stats[D + c], s2);
}

// ---------------------------------------------------------------- BN apply + leaky ReLU + pile accumulate
__global__ void k_bn_apply(float* __restrict__ acc, const float* __restrict__ stats,
                           const float* __restrict__ gamma, const float* __restrict__ beta,
                           float* __restrict__ pile) {
    int c = threadIdx.x;
    const float invN = 1.0f / (float)N_NODES;
    float mean = stats[c] * invN;
    float var  = stats[D + c] * invN - mean * mean;  // biased variance
    float sc   = gamma[c] * rsqrtf(var + BN_EPS);
    float sh   = beta[c] - mean * sc;
    long r0 = (long)blockIdx.x * 4;
#pragma unroll
    for (int i = 0; i < 4; ++i) {
        long r = r0 + i;
        if (r >= N_NODES) break;
        size_t idx = (size_t)r * D + c;
        float v = acc[idx] * sc + sh;
        v = (v >= 0.f) ? v : NEG_SLOPE * v;
        acc[idx] = v;
        pile[idx] += v;
    }
}

// ---------------------------------------------------------------- LayerNorm + residual + classifier + log_softmax
__global__ void k_final(const float* __restrict__ avg, const float* __restrict__ h,
                        const float* __restrict__ lng, const float* __restrict__ lnb,
                        const float* __restrict__ lastW, const float* __restrict__ lastb,
                        float* __restrict__ out) {
    __shared__ float red[D];
    __shared__ float sh[D];
    __shared__ float lg[N_CLASS];
    __shared__ float wlds[D * N_CLASS];      // 40 KB staged classifier weights
    int t = threadIdx.x;
    long row = blockIdx.x;

    // Kick off async staging of last_W into LDS; overlaps with the LN reductions.
#if HAVE_ASYNC_LDS
    for (int i = t * 4; i < D * N_CLASS; i += 256 * 4) {
        __builtin_amdgcn_global_load_async_to_lds_b128(
            (v4i_vs*)(lastW + i), (v4i_vs*)&wlds[i], 0, 0);
    }
#else
    for (int i = t; i < D * N_CLASS; i += 256) wlds[i] = lastW[i];
#endif

    float a = avg[row * D + t];
    red[t] = a;
    __syncthreads();
    for (int s = 128; s > 0; s >>= 1) { if (t < s) red[t] += red[t + s]; __syncthreads(); }
    float mu = red[0] * (1.0f / D);
    __syncthreads();
    float dv = a - mu;
    red[t] = dv * dv;
    __syncthreads();
    for (int s = 128; s > 0; s >>= 1) { if (t < s) red[t] += red[t + s]; __syncthreads(); }
    float var = red[0] * (1.0f / D);

    float v  = dv * rsqrtf(var + LN_EPS) * lng[t] + lnb[t];
    float hv = h[row * D + t];
    sh[t] = hv + ALPHA_MIX * (v - hv);

#if HAVE_ASYNC_LDS
#if __has_builtin(__builtin_amdgcn_s_wait_asynccnt)
    __builtin_amdgcn_s_wait_asynccnt(0);
#else
    asm volatile("s_wait_asynccnt 0" ::: "memory");
#endif
#endif
    __syncthreads();

    if (t < N_CLASS) {
        float s = lastb[t];
        for (int c = 0; c < D; ++c) s += sh[c] * wlds[c * N_CLASS + t];
        lg[t] = s;
    }
    __syncthreads();
    if (t < N_CLASS) {
        float mx = lg[0];
        for (int j = 1; j < N_CLASS; ++j) mx = fmaxf(mx, lg[j]);
        float se = 0.f;
        for (int j = 0; j < N_CLASS; ++j) se += expf(lg[j] - mx);
        out[row * N_CLASS + t] = lg[t] - mx - logf(se);
    }
}

// ---------------------------------------------------------------- host orchestration
extern "C" void kernel_launch(void* const* d_in, const int* in_sizes, int n_in,
                              void* d_out, int out_size, void* d_ws, size_t ws_size,
                              hipStream_t stream) {
    (void)in_sizes; (void)n_in; (void)out_size; (void)ws_size;
    const float* x     = (const float*)d_in[0];
    const int*   esrc  = (const int*)d_in[1];
    const int*   edst  = (const int*)d_in[2];
    const float* ew    = (const float*)d_in[3];
    const float* gW    = (const float*)d_in[4];
    const float* gb    = (const float*)d_in[5];
    const float* bng   = (const float*)d_in[6];
    const float* bnb   = (const float*)d_in[7];
    const float* avgW  = (const float*)d_in[8];
    const float* avgb  = (const float*)d_in[9];
    const float* lng   = (const float*)d_in[10];
    const float* lnb   = (const float*)d_in[11];
    const float* lastW = (const float*)d_in[12];
    const float* lastb = (const float*)d_in[13];
    float* out = (float*)d_out;

    const size_t NB = (size_t)N_NODES * D;
    float* buf0  = (float*)d_ws;       // activation ping
    float* buf1  = buf0 + NB;          // GEMM output ("support"), later avg_vec
    float* buf2  = buf1 + NB;          // activation pong
    float* pile  = buf2 + NB;          // sum of layer outputs
    float* stats = pile + NB;          // 2*D BN stats

    k_zero<<<2048, 256, 0, stream>>>(pile, (long)NB);

    const float* hin = x;
    float* acc = buf0;
    for (int i = 0; i < N_LAYERS; ++i) {
        k_gemm_wmma<<<dim3(N_NODES / 16), dim3(32, 4), 0, stream>>>(
            hin, gW + (size_t)i * D * D, buf1, nullptr, 1.0f);
        k_fill_bias<<<(N_NODES + 3) / 4, 256, 0, stream>>>(acc, gb + i * D);
        k_spmm<<<NUM_EDGES / 32, 256, 0, stream>>>(esrc, edst, ew, buf1, acc);
        k_zero<<<2, 256, 0, stream>>>(stats, 2 * D);
        k_bn_stats<<<800, 256, 0, stream>>>(acc, stats);
        k_bn_apply<<<(N_NODES + 3) / 4, 256, 0, stream>>>(
            acc, stats, bng + i * D, bnb + i * D, pile);
        hin = acc;
        acc = (acc == buf0) ? buf2 : buf0;
    }

    // avg_vec = (pile / N_LAYERS) @ avg_W + avg_b   (fold 1/4 into A-scale)
    k_gemm_wmma<<<dim3(N_NODES / 16), dim3(32, 4), 0, stream>>>(
        pile, avgW, buf1, avgb, 1.0f / (float)N_LAYERS);

    k_final<<<N_NODES, 256, 0, stream>>>(buf1, hin, lng, lnb, lastW, lastb, out);
}